// WaveletTransform_87729001988370
// MI455X (gfx1250) — compile-verified
//
#include <hip/hip_runtime.h>

// ---------------- CDNA5 (gfx1250) fused 2-D db4 DWT ----------------
// out[4][nimg][259][259] = dwt2(x[nimg][512][512]), symmetric padding.
// Phase 1 (row filter) -> LDS, Phase 2 (col filter) = 16x16x4 F32 WMMA.

typedef __attribute__((ext_vector_type(2))) float v2f;
typedef __attribute__((ext_vector_type(8))) float v8f;

#define HH_   512
#define WW_   512
#define HO_   259
#define WO_   259
#define TOH   16                    // output rows per strip
#define KROWS (2*TOH + 6)           // 38 intermediate rows needed
#define KP    40                    // padded to multiple of 4 (WMMA K)
#define WP    272                   // LDS row pitch (covers col 271 of last tile)
#define NSTRIPS ((HO_ + TOH - 1)/TOH)   // 17
#define NWT     ((WO_ + 15)/16)         // 17 wo-tiles

// db4 decomposition filters, time-reversed (cross-correlation taps):
// out[o] = sum_t K[t] * x[reflect(2*o + t - 6)]
static constexpr float KLO[8] = {
     0.23037781330885523f,  0.7148465705525415f,   0.6308807679295904f,
    -0.02798376941698385f, -0.18703481171888114f,  0.030841381835986965f,
     0.032883011666982945f, -0.010597401784997278f };
static constexpr float KHI[8] = {
    -0.010597401784997278f, -0.032883011666982945f, 0.030841381835986965f,
     0.18703481171888114f,  -0.02798376941698385f, -0.6308807679295904f,
     0.7148465705525415f,   -0.23037781330885523f };

__device__ __forceinline__ int reflect512(int i) {
    i = (i < 0) ? (-i - 1) : i;
    i = (i >= 512) ? (1023 - i) : i;
    return i;
}

// Branchless 8-way coefficient select (d outside [0,7] -> 0).
__device__ __forceinline__ float pick8(int d, const float c[8]) {
    float r = 0.0f;
    r = (d == 0) ? c[0] : r;
    r = (d == 1) ? c[1] : r;
    r = (d == 2) ? c[2] : r;
    r = (d == 3) ? c[3] : r;
    r = (d == 4) ? c[4] : r;
    r = (d == 5) ? c[5] : r;
    r = (d == 6) ? c[6] : r;
    r = (d == 7) ? c[7] : r;
    return r;
}

__global__ __launch_bounds__(256)
void dwt2_db4_wmma_kernel(const float* __restrict__ x,
                          float* __restrict__ out,
                          int nimg) {
    extern __shared__ float smem[];
    float* slo = smem;              // [KP][WP]
    float* shi = smem + KP * WP;    // [KP][WP]

    const int tid   = threadIdx.x;
    const int img   = blockIdx.x / NSTRIPS;
    const int strip = blockIdx.x % NSTRIPS;
    const int ho0   = strip * TOH;

    const float* __restrict__ xim = x + (size_t)img * ((size_t)HH_ * WW_);

    // ---------------- Phase 1: row DWT (along W) into LDS ----------------
    for (int idx = tid; idx < KP * WP; idx += 256) {
        const int r  = idx / WP;
        const int wo = idx - r * WP;
        float lo = 0.0f, hi = 0.0f;
        if (r < KROWS && wo < WO_) {
            const int iy = reflect512(2 * ho0 - 6 + r);
            const float* __restrict__ row = xim + (size_t)iy * WW_;
            const int base = 2 * wo - 6;
            if (base >= 0 && base + 7 < WW_) {
#pragma unroll
                for (int t = 0; t < 8; ++t) {
                    const float v = row[base + t];
                    lo = fmaf(KLO[t], v, lo);
                    hi = fmaf(KHI[t], v, hi);
                }
            } else {
#pragma unroll
                for (int t = 0; t < 8; ++t) {
                    const float v = row[reflect512(base + t)];
                    lo = fmaf(KLO[t], v, lo);
                    hi = fmaf(KHI[t], v, hi);
                }
            }
        }
        slo[idx] = lo;   // zero-pads rows [KROWS,KP) and cols [WO_,WP)
        shi[idx] = hi;
    }
    __syncthreads();

    // ---------------- Phase 2: column DWT (along H) via WMMA ----------------
    // Y(16x16) = F(16x40) * X(40x16), F[m][k] = filt[k - 2m] banded matrix.
    const int lane  = tid & 31;
    const int wave  = tid >> 5;
    const int m     = lane & 15;            // A-matrix row / D-matrix N index
    const int krsel = (lane >> 4) << 1;     // K selector: lanes 0-15 -> 0, 16-31 -> 2

    // A-fragments depend only on (lane, kk): precompute once per wave.
    v2f aLo[KP / 4], aHi[KP / 4];
#pragma unroll
    for (int kk = 0; kk < KP / 4; ++kk) {
        const int d = 4 * kk + krsel - 2 * m;
        aLo[kk].x = pick8(d,     KLO);
        aLo[kk].y = pick8(d + 1, KLO);
        aHi[kk].x = pick8(d,     KHI);
        aHi[kk].y = pick8(d + 1, KHI);
    }

    const size_t plane_stride = (size_t)nimg * HO_ * WO_;

    for (int wt = wave; wt < NWT; wt += 8) {
        const int wo0 = wt * 16;
        const int col = wo0 + m;            // B-matrix N index for this lane

        v8f aLL = {}, aLH = {}, aHL = {}, aHH = {};
#pragma unroll
        for (int kk = 0; kk < KP / 4; ++kk) {
            const int kb = 4 * kk + krsel;
            v2f bl, bh;
            bl.x = slo[kb * WP + col];
            bl.y = slo[(kb + 1) * WP + col];
            bh.x = shi[kb * WP + col];
            bh.y = shi[(kb + 1) * WP + col];

            aLL = __builtin_amdgcn_wmma_f32_16x16x4_f32(false, aLo[kk], false, bl,
                                                        (short)0, aLL, false, false);
            aLH = __builtin_amdgcn_wmma_f32_16x16x4_f32(false, aHi[kk], false, bl,
                                                        (short)0, aLH, false, false);
            aHL = __builtin_amdgcn_wmma_f32_16x16x4_f32(false, aLo[kk], false, bh,
                                                        (short)0, aHL, false, false);
            aHH = __builtin_amdgcn_wmma_f32_16x16x4_f32(false, aHi[kk], false, bh,
                                                        (short)0, aHH, false, false);
        }

        // D layout: lane L, reg v -> (M = v + 8*(L>=16), N = L&15)
        const int wos = wo0 + m;
        if (wos < WO_) {
#pragma unroll
            for (int v = 0; v < 8; ++v) {
                const int mrow = v + ((lane >> 4) << 3);
                const int ho   = ho0 + mrow;
                if (ho < HO_) {
                    const size_t off = ((size_t)img * HO_ + ho) * WO_ + wos;
                    out[0 * plane_stride + off] = aLL[v];
                    out[1 * plane_stride + off] = aLH[v];
                    out[2 * plane_stride + off] = aHL[v];
                    out[3 * plane_stride + off] = aHH[v];
                }
            }
        }
    }
}

extern "C" void kernel_launch(void* const* d_in, const int* in_sizes, int n_in,
                              void* d_out, int out_size, void* d_ws, size_t ws_size,
                              hipStream_t stream) {
    const float* x = (const float*)d_in[0];
    float* out = (float*)d_out;
    const int nimg = in_sizes[0] / (HH_ * WW_);   // 8*32 = 256

    const dim3 grid((unsigned)(nimg * NSTRIPS));
    const dim3 block(256);
    const size_t shmem = (size_t)2 * KP * WP * sizeof(float);  // 87040 B

    dwt2_db4_wmma_kernel<<<grid, block, shmem, stream>>>(x, out, nimg);
}